// CUDARealRayTracer_50379966382329
// MI455X (gfx1250) — compile-verified
//
#include <hip/hip_runtime.h>
#include <cmath>

// Problem constants (match reference)
#define IMG_H 512
#define IMG_W 512
#define NSPEC 4
#define STEP_SIZE 0.05f
#define SOURCE_SPREAD 0.1f
#define NBLOCKS 8          // MARCH_STEPS / NOISE_EVERY
#define NOISE_EVERY 8

// Hot-window staging: all samples land at image center (ix = px + 255.5,
// |px| <~ 2.5).  Stage a 32x32 window [240,272) into LDS.
#define WB  240
#define WSZ 32

__global__ __launch_bounds__(256)
void raymarch_kernel(const float* __restrict__ vol,           // [B,1,H,W]
                     const float* __restrict__ origin_noise,  // [B,R,2]
                     const float* __restrict__ scatter_noise, // [4,8,B,R]
                     const float* __restrict__ ray_intensity, // [4]
                     const float* __restrict__ tds_scale,     // [1]
                     const float* __restrict__ scat_coef,     // [4]
                     float* __restrict__ out,                 // [B,4,gs,gs]
                     int B, int R, int gs)
{
    __shared__ float patch[WSZ * WSZ];

    const int blocksPerBatch = R >> 8;                 // 256 rays per block
    const int b = blockIdx.x / blocksPerBatch;
    const int r = (blockIdx.x % blocksPerBatch) * 256 + threadIdx.x;

    const float* gvol = vol + (size_t)b * (IMG_H * IMG_W);

    // ---- CDNA5 async DMA: stage 32x32 f32 window (4KB) global -> LDS ----
    {
        const int t   = threadIdx.x;          // 256 chunks x 16B = 4KB
        const int row = t >> 3;               // 0..31
        const int c4  = (t & 7) << 2;         // 0,4,...,28 (floats)
        unsigned ldsOff = (unsigned)(size_t)&patch[(row << 5) + c4];
        unsigned gOff   = (unsigned)((((WB + row) * IMG_W) + WB + c4) * sizeof(float));
        asm volatile("global_load_async_to_lds_b128 %0, %1, %2"
                     :: "v"(ldsOff), "v"(gOff), "s"(gvol) : "memory");
        asm volatile("s_wait_asynccnt 0" ::: "memory");
    }
    __syncthreads();

    // ---- spectral constants (wl^-1.5, wl^-4 precomputed at f64, rounded) ----
    const float wlm15[NSPEC] = {44.19417382f, 24.05626122f, 13.09457213f, 8.0f};
    const float wlm4 [NSPEC] = {24414.0625f, 4822.530864f, 952.5986883f, 256.0f};
    const float tds = tds_scale[0];
    float total[NSPEC], nscale[NSPEC], inten[NSPEC];
#pragma unroll
    for (int s = 0; s < NSPEC; ++s) {
        const float sc = scat_coef[s] * wlm4[s];     // scattering term
        total[s]  = tds * wlm15[s] + sc;             // absorb + scatter
        nscale[s] = sc * 0.01f;
        inten[s]  = ray_intensity[s];
    }

    // ---- ray setup ----
    const float ox = origin_noise[((size_t)b * R + r) * 2 + 0] * SOURCE_SPREAD;
    const float oy = origin_noise[((size_t)b * R + r) * 2 + 1] * SOURCE_SPREAD;
    const float fx = (float)(r % gs) / (float)(gs - 1) * 2.0f - 1.0f;
    const float fy = (float)(r / gs) / (float)(gs - 1) * 2.0f - 1.0f;
    const float dx = fx * 1.5f, dy = fy * 1.5f;
    const float inv = 1.0f / sqrtf(dx * dx + dy * dy + 4.0f);  // dz = -2
    const float dirx = dx * inv, diry = dy * inv;

    const size_t BR = (size_t)B * R;
    const float* nbase = scatter_noise + (size_t)b * R + r;

    // ---- march: 8 blocks of 8 steps; 2 exps per block per wavelength ----
#pragma unroll 1
    for (int j = 0; j < NBLOCKS; ++j) {
        float d0 = 0.0f, ssum = 0.0f;
#pragma unroll
        for (int k = 0; k < NOISE_EVERY; ++k) {
            const float tt = (float)(j * NOISE_EVERY + k) * STEP_SIZE;
            const float px = fmaf(dirx, tt, ox);
            const float py = fmaf(diry, tt, oy);
            // normalized coords, clipped, then unnormalized + border clip
            const float gxn = fminf(fmaxf(px * (2.0f / IMG_W), -1.0f), 1.0f);
            const float gyn = fminf(fmaxf(py * (2.0f / IMG_H), -1.0f), 1.0f);
            const float ix = fminf(fmaxf(fmaf(gxn + 1.0f, IMG_W * 0.5f, -0.5f),
                                         0.0f), (float)(IMG_W - 1));
            const float iy = fminf(fmaxf(fmaf(gyn + 1.0f, IMG_H * 0.5f, -0.5f),
                                         0.0f), (float)(IMG_H - 1));
            const float x0f = floorf(ix), y0f = floorf(iy);
            const float wx = ix - x0f, wy = iy - y0f;
            const int x0 = (int)x0f, y0 = (int)y0f;

            float d;
            if (((unsigned)(x0 - WB) <= (WSZ - 2)) &
                ((unsigned)(y0 - WB) <= (WSZ - 2))) {
                // hot path: bilinear taps from the LDS window.
                const float* p = &patch[(y0 - WB) * WSZ + (x0 - WB)];
                float v00 = p[0], v01 = p[1], v10 = p[WSZ], v11 = p[WSZ + 1];
                // Pin the loads inside this branch so SimplifyCFG cannot
                // sink/merge them with the global fallback loads (which would
                // force a generic-pointer phi and lower to flat_load instead
                // of ds_load).
                asm volatile("" : "+v"(v00), "+v"(v01), "+v"(v10), "+v"(v11));
                d = (v00 * (1.0f - wx) + v01 * wx) * (1.0f - wy)
                  + (v10 * (1.0f - wx) + v11 * wx) * wy;
            } else {
                // cold fallback: full border-clamped global gather
                const int x1 = min(x0 + 1, IMG_W - 1);
                const int y1 = min(y0 + 1, IMG_H - 1);
                float v00 = gvol[y0 * IMG_W + x0];
                float v01 = gvol[y0 * IMG_W + x1];
                float v10 = gvol[y1 * IMG_W + x0];
                float v11 = gvol[y1 * IMG_W + x1];
                asm volatile("" : "+v"(v00), "+v"(v01), "+v"(v10), "+v"(v11));
                d = (v00 * (1.0f - wx) + v01 * wx) * (1.0f - wy)
                  + (v10 * (1.0f - wx) + v11 * wx) * wy;
            }
            if (k == 0) d0 = d; else ssum += d;
        }
        const float e0 = -STEP_SIZE * d0;    // log of att[j,0] / total
        const float es = -STEP_SIZE * ssum;  // log of prod att[j,1:] / total
#pragma unroll
        for (int s = 0; s < NSPEC; ++s) {
            const float noise = nbase[(size_t)(s * NBLOCKS + j) * BR] * nscale[s];
            const float a0 = __expf(total[s] * e0);
            const float ap = __expf(total[s] * es);
            inten[s] = fmaxf(fmaf(inten[s], a0, noise), 0.0f) * ap;
        }
    }

    // ---- output: [B, 4, gs, gs] -> flat b*4*R + s*R + r ----
    float* obase = out + (size_t)b * NSPEC * R + r;
#pragma unroll
    for (int s = 0; s < NSPEC; ++s) obase[(size_t)s * R] = inten[s];
}

extern "C" void kernel_launch(void* const* d_in, const int* in_sizes, int n_in,
                              void* d_out, int out_size, void* d_ws, size_t ws_size,
                              hipStream_t stream) {
    const float* vol           = (const float*)d_in[0];
    const float* origin_noise  = (const float*)d_in[1];
    const float* scatter_noise = (const float*)d_in[2];
    const float* ray_intensity = (const float*)d_in[3];
    const float* tds           = (const float*)d_in[4];
    const float* sc            = (const float*)d_in[5];

    const int B = in_sizes[0] / (IMG_H * IMG_W);
    const int R = in_sizes[1] / (B * 2);
    const int gs = (int)(sqrt((double)R) + 0.5);

    const int blocks = B * (R / 256);
    raymarch_kernel<<<blocks, 256, 0, stream>>>(
        vol, origin_noise, scatter_noise, ray_intensity, tds, sc,
        (float*)d_out, B, R, gs);
}